// BipolarLIFNeuron_13726715478518
// MI455X (gfx1250) — compile-verified
//
#include <hip/hip_runtime.h>
#include <cstdint>

// Bipolar LIF forward: out[b,t,n,{f, F+f}] = {pos, neg} spikes.
// B=8, T=32, N=1024, F=128. Memory-bound: 128MB in + 256MB out -> ~16.5us
// floor @ 23.3 TB/s. CDNA5 path: GLOBAL_LOAD_ASYNC_TO_LDS_B128 (NT) double
// buffer + s_wait_asynccnt pipelining; NT B128 spike stores.

#define LIF_BLOCK 256
#define LIF_VEC   4
#define LIF_TILE  (LIF_BLOCK * LIF_VEC)   // 1024 floats / block / timestep

typedef float vfloat4 __attribute__((ext_vector_type(4)));

namespace lifcfg {
constexpr int B = 8, T = 32, N = 1024, F = 128;
constexpr int C = N * F;                  // 131072 channels per batch
constexpr int TILES_PER_B = C / LIF_TILE; // 128
}

__device__ __forceinline__ void async_copy_b128(uint32_t lds_byte_addr,
                                                const float* gptr) {
  // copies 16B/lane from the global address in vaddr to the LDS address in vdst, tracked by ASYNCcnt; NT: stream once.
  asm volatile("global_load_async_to_lds_b128 %0, %1, off th:TH_LOAD_NT"
               :
               : "v"(lds_byte_addr), "v"(gptr)
               : "memory");
}
__device__ __forceinline__ void wait_async_le1() {
  asm volatile("s_wait_asynccnt 1" ::: "memory");
}
__device__ __forceinline__ void wait_async_le0() {
  asm volatile("s_wait_asynccnt 0" ::: "memory");
}
__device__ __forceinline__ void wait_ds0() {
  // Ensure the ds_load_b128 read-back completed before the async engine may
  // overwrite the staging buffer (async LDS writes are unordered vs DS ops).
  asm volatile("s_wait_dscnt 0" ::: "memory");
}

__device__ __forceinline__ void lif_step(const vfloat4 xv, float V[LIF_VEC],
                                         float* orow) {
  const float alpha = 0.95122942450071400909f;  // exp(-1/20)
  float pos[LIF_VEC], neg[LIF_VEC];
#pragma unroll
  for (int j = 0; j < LIF_VEC; ++j) {
    float v = fmaf(alpha, V[j], xv[j]);         // V = a*V + x
    pos[j] = (v >= 1.0f) ? 1.0f : 0.0f;         // spike_pos
    neg[j] = (v <= -1.0f) ? 1.0f : 0.0f;        // spike_neg
    V[j] = v - pos[j] - neg[j];                 // V -= (pos*tp - neg*tn)
  }
  vfloat4 pv = {pos[0], pos[1], pos[2], pos[3]};
  vfloat4 nv = {neg[0], neg[1], neg[2], neg[3]};
  __builtin_nontemporal_store(pv, reinterpret_cast<vfloat4*>(orow));
  __builtin_nontemporal_store(nv, reinterpret_cast<vfloat4*>(orow + lifcfg::F));
}

__global__ __launch_bounds__(LIF_BLOCK, 2)
void bipolar_lif_fwd(const float* __restrict__ x, float* __restrict__ out) {
  using namespace lifcfg;
  __shared__ __align__(16) float stage[2][LIF_TILE];

  const int tid = threadIdx.x;
  const int blk = blockIdx.x;                 // 0 .. B*TILES_PER_B-1
  const int b   = blk / TILES_PER_B;
  const int ct  = blk % TILES_PER_B;
  const int c   = ct * LIF_TILE + tid * LIF_VEC;  // channel index (mult of 4)
  const int n   = c / F;
  const int f0  = c % F;

  // Input: x[(b*T + t)*C + c]; +C floats per t.
  const float* gs = x + (size_t)b * T * C + c;
  // Output: out[((b*T + t)*N + n)*2F + f0] (pos), +F (neg); +OS per t.
  float* orow = out + ((size_t)(b * T) * N + n) * (2 * F) + f0;
  const size_t OS = (size_t)N * 2 * F;

  const uint32_t lds0 = (uint32_t)(uintptr_t)&stage[0][tid * LIF_VEC];
  const uint32_t lds1 = (uint32_t)(uintptr_t)&stage[1][tid * LIF_VEC];
  const vfloat4* sp0 = reinterpret_cast<const vfloat4*>(&stage[0][tid * LIF_VEC]);
  const vfloat4* sp1 = reinterpret_cast<const vfloat4*>(&stage[1][tid * LIF_VEC]);

  float V[LIF_VEC] = {0.f, 0.f, 0.f, 0.f};

  async_copy_b128(lds0, gs);                       // prime: t=0 -> buf0

  // Steady state: pairs t = (2tt, 2tt+1), tt = 0..14  (t = 0..29)
  for (int tt = 0; tt < T / 2 - 1; ++tt) {
    async_copy_b128(lds1, gs + C);                 // t+1 -> buf1
    wait_async_le1();                              // buf0 ready (in-order)
    const vfloat4 xv0 = *sp0;
    wait_ds0();                                    // buf0 fully read
    async_copy_b128(lds0, gs + 2 * (size_t)C);     // t+2 -> buf0
    lif_step(xv0, V, orow);                        // even step

    wait_async_le1();                              // buf1 ready
    const vfloat4 xv1 = *sp1;
    wait_ds0();                                    // buf1 fully read
    lif_step(xv1, V, orow + OS);                   // odd step

    gs   += 2 * (size_t)C;
    orow += 2 * OS;
  }

  // Tail: t = T-2, T-1 (buf0 load for T-2 already in flight)
  async_copy_b128(lds1, gs + C);                   // t=T-1 -> buf1
  wait_async_le1();
  const vfloat4 xv0 = *sp0;
  wait_ds0();
  lif_step(xv0, V, orow);
  wait_async_le0();
  const vfloat4 xv1 = *sp1;
  lif_step(xv1, V, orow + OS);
}

extern "C" void kernel_launch(void* const* d_in, const int* in_sizes, int n_in,
                              void* d_out, int out_size, void* d_ws, size_t ws_size,
                              hipStream_t stream) {
  using namespace lifcfg;
  (void)in_sizes; (void)n_in; (void)out_size; (void)d_ws; (void)ws_size;
  const float* x = (const float*)d_in[0];
  float* out = (float*)d_out;
  const dim3 grid(B * TILES_PER_B);   // 1024 blocks
  const dim3 block(LIF_BLOCK);        // 256 threads = 8 wave32
  hipLaunchKernelGGL(bipolar_lif_fwd, grid, block, 0, stream, x, out);
}